// FusedPointwise_69879117906373
// MI455X (gfx1250) — compile-verified
//
#include <hip/hip_runtime.h>
#include <hip/hip_bf16.h>

// MI455X (gfx1250) fused grouped 1x1 conv + BN + ReLU + residual.
// Memory-bound: ~1.45 GFLOP vs ~145 MB traffic (AI ~10 FLOP/B) -> ~6.2us
// floor at 23.3 TB/s HBM; fp32 V_WMMA_16X16X4 easily covers the math.
// Strategy: per-(n,g) workgroup stages scale-folded, zero-padded weights in
// LDS once; 8 waves stride the 49 P-tiles; hot loop = coalesced global x
// loads + conflict-free ds_load_b64 A-frags + WMMA; epilogue = +bias (regs),
// ReLU, +residual, store. Only x/residual/out touch global memory.

typedef __attribute__((ext_vector_type(2))) float v2f;
typedef __attribute__((ext_vector_type(8))) float v8f;

namespace {
constexpr int NB     = 32;    // batch
constexpr int CIN    = 480;
constexpr int COUT   = 480;
constexpr int GROUPS = 8;
constexpr int KG     = 60;    // C_in per group (K)
constexpr int MG     = 60;    // C_out per group (M)
constexpr int MPAD   = 64;    // M padded to 4 WMMA tiles (rows 60..63 zero)
constexpr int HW     = 784;   // 28*28 (P)
constexpr int PTILES = HW / 16;            // 49 column tiles of 16
constexpr int WAVES_PER_BLOCK = 8;         // 256 threads, wave32
}

__global__ __launch_bounds__(256)
void conv1x1g_bn_relu_res_wmma(const float* __restrict__ x,
                               const float* __restrict__ w,     // [COUT][KG]
                               const float* __restrict__ scale, // [COUT]
                               const float* __restrict__ bias,  // [COUT]
                               const float* __restrict__ res,   // [NB][COUT][HW]
                               float* __restrict__ out)         // [NB][COUT][HW]
{
    __shared__ float wlds[MPAD * KG];      // 64x60 fp32, scale-folded, 15360 B

    const int lane = threadIdx.x & 31;
    const int wave = threadIdx.x >> 5;

    const int ng = blockIdx.x;             // one block per (n, g)
    const int g  = ng % GROUPS;
    const int n  = ng / GROUPS;

    // ---- Stage W' = W * scale (row-wise) into LDS, zero-pad rows 60..63.
    for (int i = threadIdx.x; i < MPAD * KG; i += 256) {
        const int row = i / KG;
        const int col = i - row * KG;
        float v = 0.0f;
        if (row < MG) {
            const int c = g * MG + row;
            v = w[(size_t)c * KG + col] * scale[c];
        }
        wlds[i] = v;
    }
    __syncthreads();

    const int half = lane >> 4;            // 0: lanes 0-15, 1: lanes 16-31
    const int l    = lane & 15;

    // ---- Per-lane bias registers for the C/D layout:
    // VGPR r of tile t holds row m = t*16 + half*8 + r, col = pbase + l.
    float bi[4][8];
    #pragma unroll
    for (int t = 0; t < 4; ++t) {
        #pragma unroll
        for (int r = 0; r < 8; ++r) {
            const int m = t * 16 + half * 8 + r;
            bi[t][r] = (m < MG) ? bias[g * MG + m] : 0.0f;
        }
    }

    // Base pointers for this (n, g)
    const float* __restrict__ xg = x + ((size_t)n * CIN + (size_t)g * KG) * HW;
    const size_t cg_base = ((size_t)n * COUT + (size_t)g * MG) * HW;

    // Per-lane LDS pointer for A-fragments: row l (+16t via imm), K offset kb.
    // Byte addr = l*240 + half*8 + t*3840 + k0*4 -> always 8B aligned.
    const float* __restrict__ wl = wlds + l * KG + half * 2;

    // ---- 8 waves stride the 49 P-tiles of this (n, g).
    for (int pt = wave; pt < PTILES; pt += WAVES_PER_BLOCK) {
        const int pbase = pt * 16;
        const int p     = pbase + l;

        // Prefetch this tile's residual rows (0..59 covered by the two ranges)
        // so the epilogue stream overlaps the K loop.
        __builtin_prefetch(res + cg_base + (size_t)lane * HW + pbase, 0, 1);
        __builtin_prefetch(res + cg_base + (size_t)(59 - lane) * HW + pbase, 0, 1);

        v8f acc[4] = {};
        const float* __restrict__ xb = xg + p;   // per-lane column pointer

        #pragma unroll 5
        for (int k0 = 0; k0 < KG; k0 += 4) {
            // B fragment (4x16 KxP), f32 layout:
            //   VGPR0: lanes 0-15 -> K=k0,   lanes 16-31 -> K=k0+2
            //   VGPR1: lanes 0-15 -> K=k0+1, lanes 16-31 -> K=k0+3
            const int kb = k0 + half * 2;
            v2f b;
            b.x = xb[(size_t)kb * HW];
            b.y = xb[(size_t)(kb + 1) * HW];

            // A fragments from LDS (unconditional; pad rows are zero).
            #pragma unroll
            for (int t = 0; t < 4; ++t) {
                const v2f a = *(const v2f*)(wl + t * 16 * KG + k0);
                acc[t] = __builtin_amdgcn_wmma_f32_16x16x4_f32(
                    /*neg_a=*/false, a, /*neg_b=*/false, b,
                    /*c_mod=*/(short)0, acc[t],
                    /*reuse_a=*/false, /*reuse_b=*/false);
            }
        }

        // ---- Epilogue: y = relu(acc + bias) + residual   (scale pre-folded)
        #pragma unroll
        for (int t = 0; t < 4; ++t) {
            #pragma unroll
            for (int r = 0; r < 8; ++r) {
                const int m = t * 16 + half * 8 + r;
                if (m < MG) {
                    float v = fmaxf(acc[t][r] + bi[t][r], 0.0f);
                    const size_t idx = cg_base + (size_t)m * HW + p;
                    out[idx] = v + res[idx];
                }
            }
        }
    }
}

extern "C" void kernel_launch(void* const* d_in, const int* in_sizes, int n_in,
                              void* d_out, int out_size, void* d_ws, size_t ws_size,
                              hipStream_t stream) {
    (void)in_sizes; (void)n_in; (void)d_ws; (void)ws_size; (void)out_size;
    const float* x     = (const float*)d_in[0];
    const float* w     = (const float*)d_in[1];
    const float* scale = (const float*)d_in[2];
    const float* bias  = (const float*)d_in[3];
    const float* res   = (const float*)d_in[4];
    // d_in[5] is `groups` (scalar, ==8) — baked into the kernel constants.
    float* out = (float*)d_out;

    conv1x1g_bn_relu_res_wmma<<<NB * GROUPS, 256, 0, stream>>>(
        x, w, scale, bias, res, out);
}